// PyTorchDWT2D_22110491640208
// MI455X (gfx1250) — compile-verified
//
#include <hip/hip_runtime.h>

// ---------------------------------------------------------------------------
// 2-D Haar DWT, level 1, fp32, shape (16,1,2048,2048) -> 4 x (16,1,1024,1024).
//
// Memory-bound: 512 MiB total traffic -> ~23 us floor at 23.3 TB/s HBM.
// Strategy: streaming kernel, all-b128 NT loads/stores, gfx1250
// global_prefetch_b8 for extra MLP. WMMA deliberately NOT used: arithmetic
// intensity is ~0.5 FLOP/byte, far below the matrix-op roofline; a matrix
// formulation only adds lane-shuffle work on the bandwidth-critical path.
//
// Each thread: 8 input columns x 2 input rows (4 x b128 NT loads, 64 B)
//           -> 4 output columns for each of 4 subbands (4 x b128 NT stores).
// ---------------------------------------------------------------------------

typedef float v4f __attribute__((ext_vector_type(4)));

__device__ __forceinline__ float safef(float v) {
    // match jnp.where(isfinite(x), x, 0)
    return __builtin_isfinite(v) ? v : 0.0f;
}

// Problem constants (fixed by the reference setup_inputs()).
constexpr int BATCH = 16;
constexpr int H     = 2048;
constexpr int W     = 2048;
constexpr int H2    = H / 2;   // 1024
constexpr int W2    = W / 2;   // 1024
constexpr long long N4 = (long long)BATCH * H2 * W2;  // elements per subband (16M)

__global__ __launch_bounds__(256) void dwt2d_haar_kernel(
    const float* __restrict__ x, float* __restrict__ out)
{
    const int tid = blockIdx.x * blockDim.x + threadIdx.x;   // 0 .. N4/4-1

    // 256 groups of 8 input columns per output row. Consecutive tids map to
    // 32B-strided addresses; the two b128 loads per row jointly cover a
    // contiguous 1KB span per wave -> every cache line fully requested.
    const int r  = tid >> 8;          // flat output row: 0 .. 16*1024-1
    const int g  = tid & 255;         // column group within the row
    const int b  = r >> 10;           // batch
    const int h2 = r & 1023;          // output row within image

    const float* rowTop = x + (size_t)b * H * W + (size_t)(2 * h2) * W + (size_t)g * 8;
    const float* rowBot = rowTop + W;

    // Streaming NT loads: data is read exactly once; 512 MiB footprint
    // exceeds the 192 MB L2.
    v4f t0 = __builtin_nontemporal_load((const v4f*)rowTop);
    v4f t1 = __builtin_nontemporal_load((const v4f*)(rowTop + 4));
    v4f u0 = __builtin_nontemporal_load((const v4f*)rowBot);
    v4f u1 = __builtin_nontemporal_load((const v4f*)(rowBot + 4));

    // gfx1250 prefetch path (global_prefetch_b8): pull lines 8 input rows
    // (64 KB) ahead. 4 lanes span 128 B -> one prefetch per 4 lanes covers
    // every line. Speculative prefetch past the buffer end is dropped.
    if ((tid & 3) == 0) {
        __builtin_prefetch(rowTop + 8 * W, 0, 0);
        __builtin_prefetch(rowBot + 8 * W, 0, 0);
    }

    const float top[8] = { t0.x, t0.y, t0.z, t0.w, t1.x, t1.y, t1.z, t1.w };
    const float bot[8] = { u0.x, u0.y, u0.z, u0.w, u1.x, u1.y, u1.z, u1.w };

    v4f ll, lh, hl, hh;
#pragma unroll
    for (int j = 0; j < 4; ++j) {
        // _safe on inputs
        const float a = safef(top[2 * j]);
        const float bb = safef(top[2 * j + 1]);
        const float c = safef(bot[2 * j]);
        const float d = safef(bot[2 * j + 1]);
        // Haar butterfly:
        //   ll = .5((a+d)+(b+c))   hh = .5((a+d)-(b+c))
        //   lh = .5((d-a)+(b-c))   hl = .5((d-a)-(b-c))
        const float pj = a + d, qj = bb + c, rj = d - a, sj = bb - c;
        // _safe on outputs (reference applies it to each subband too)
        ll[j] = safef(0.5f * (pj + qj));
        hh[j] = safef(0.5f * (pj - qj));
        lh[j] = safef(0.5f * (rj + sj));
        hl[j] = safef(0.5f * (rj - sj));
    }

    // Output order in d_out: ll | lh | hl | hh, each N4 elements.
    float* o = out + (size_t)r * W2 + (size_t)4 * g;
    __builtin_nontemporal_store(ll, (v4f*)(o));
    __builtin_nontemporal_store(lh, (v4f*)(o + N4));
    __builtin_nontemporal_store(hl, (v4f*)(o + 2 * N4));
    __builtin_nontemporal_store(hh, (v4f*)(o + 3 * N4));
}

extern "C" void kernel_launch(void* const* d_in, const int* in_sizes, int n_in,
                              void* d_out, int out_size, void* d_ws, size_t ws_size,
                              hipStream_t stream) {
    const float* x  = (const float*)d_in[0];
    float* out      = (float*)d_out;

    // N4/4 threads total: 4,194,304 -> 16384 blocks of 256 (8 wave32 waves).
    const int threads = 256;
    const long long total = N4 / 4;
    const int blocks = (int)((total + threads - 1) / threads);

    dwt2d_haar_kernel<<<blocks, threads, 0, stream>>>(x, out);
}